// DensityFieldLinear_62208306315803
// MI455X (gfx1250) — compile-verified
//
#include <hip/hip_runtime.h>

typedef float v2f __attribute__((ext_vector_type(2)));
typedef float v8f __attribute__((ext_vector_type(8)));

#define IN_F   4096
#define OUT_F  16384
#define M_ROWS 64
#define KK     16
#define NUM_BANKS 64
#define ANNEAL 2000
#define KC     256          // K-chunk staged in LDS: 64*256*4 = 64 KB
#define NW     8            // waves per block (wave32)
#define NTILE  (NW * 16)    // 128 output features per block
#define KSPLIT 4            // K-dimension split for occupancy
#define KSLICE (IN_F / KSPLIT)   // 1024 K per slice

__global__ __launch_bounds__(256)
void zero_out(float* __restrict__ out, int n4)
{
    int i = blockIdx.x * blockDim.x + threadIdx.x;
    if (i < n4) ((float4*)out)[i] = make_float4(0.f, 0.f, 0.f, 0.f);
}

__global__ __launch_bounds__(256)
void density_linear_wmma(const float* __restrict__ x,
                         const float* __restrict__ lw,
                         const float* __restrict__ scale,
                         const float* __restrict__ thr,
                         const int*   __restrict__ stepp,
                         float* __restrict__ out)
{
    __shared__ float xs[M_ROWS * KC];

    const int tid   = threadIdx.x;
    const int lane  = tid & 31;
    const int wave  = tid >> 5;
    const int nbase = blockIdx.x * NTILE + wave * 16;
    const int ncol  = nbase + (lane & 15);
    const int koff  = (lane >> 4) * 2;     // lanes 0-15 -> K=0,1 ; lanes 16-31 -> K=2,3
    const int kbase = blockIdx.y * KSLICE; // this block's K slice

    // step is a device scalar
    const int step = stepp[0];
    float a = fminf((float)step / (float)ANNEAL, 1.0f);
    const float oma = 1.0f - a;
    const float a8  = a * 0.125f;          // alpha/8 since 2*q-1 = cnt/8 - 1

    // uniform sorted threshold bank (16 entries) -> scalar registers
    const float* Rp = thr + (unsigned)(step % NUM_BANKS) * KK;
    float r[KK];
    #pragma unroll
    for (int i = 0; i < KK; ++i) r[i] = Rp[i];

    const float s = scale[ncol];

    v8f acc[4];
    #pragma unroll
    for (int mt = 0; mt < 4; ++mt) acc[mt] = (v8f)0.0f;

    const float* wrow = lw + (size_t)ncol * IN_F + koff;

    for (int kc = kbase; kc < kbase + KSLICE; kc += KC) {
        // ---- stage x[:, kc:kc+KC] into LDS, row-major [64][KC] ----
        __syncthreads();
        for (int i = tid * 4; i < M_ROWS * KC; i += 256 * 4) {
            const int m = i / KC;
            const int j = i & (KC - 1);
            const float4 v = *(const float4*)(x + (size_t)m * IN_F + kc + j);
            *(float4*)(xs + i) = v;
        }
        __syncthreads();

        // hint the next weight chunk into cache while we chew on this one
        if (kc + KC < kbase + KSLICE)
            __builtin_prefetch(wrow + kc + KC, 0, 1);

        #pragma unroll 4
        for (int k = 0; k < KC; k += 4) {
            // ---- B tile: quantize 2 weight elements per lane (4K x 16N) ----
            const float2 wv = *(const float2*)(wrow + kc + k);
            v2f b;
            {
                float c0 = fminf(fmaxf(wv.x, -1.0f), 1.0f);
                float c1 = fminf(fmaxf(wv.y, -1.0f), 1.0f);
                float t0 = __builtin_fmaf(c0, 0.5f, 0.5f);  // target density
                float t1 = __builtin_fmaf(c1, 0.5f, 0.5f);
                int n0 = 0, n1 = 0;
                #pragma unroll
                for (int i = 0; i < KK; ++i) {              // searchsorted(side='left')
                    n0 += (r[i] < t0) ? 1 : 0;
                    n1 += (r[i] < t1) ? 1 : 0;
                }
                // eff = (1-a)*clamped + a*(2*q - 1),  q = cnt/16
                b.x = __builtin_fmaf(oma, c0, __builtin_fmaf((float)n0, a8, -a));
                b.y = __builtin_fmaf(oma, c1, __builtin_fmaf((float)n1, a8, -a));
            }

            // ---- 4 M-tiles reuse the same B operand ----
            #pragma unroll
            for (int mt = 0; mt < 4; ++mt) {
                const int m = mt * 16 + (lane & 15);
                v2f av = *(const v2f*)(xs + m * KC + (k & (KC - 1)) + koff);
                acc[mt] = __builtin_amdgcn_wmma_f32_16x16x4_f32(
                    false, av, false, b, (short)0, acc[mt], false, false);
            }
        }
    }

    // ---- accumulate partial D into out with HW f32 atomics ----
    // D layout: col = lane&15, row = j + 8*(lane>=16)
    const int mrow0 = (lane >> 4) * 8;
    #pragma unroll
    for (int mt = 0; mt < 4; ++mt) {
        #pragma unroll
        for (int j = 0; j < 8; ++j) {
            const int m = mt * 16 + mrow0 + j;
            unsafeAtomicAdd(&out[(size_t)m * OUT_F + ncol], acc[mt][j] * s);
        }
    }
}

extern "C" void kernel_launch(void* const* d_in, const int* in_sizes, int n_in,
                              void* d_out, int out_size, void* d_ws, size_t ws_size,
                              hipStream_t stream) {
    const float* x     = (const float*)d_in[0];
    const float* lw    = (const float*)d_in[1];
    const float* scale = (const float*)d_in[2];
    const float* thr   = (const float*)d_in[3];
    const int*   step  = (const int*)d_in[4];
    float* out = (float*)d_out;

    // out_size = 64*16384 floats; zero it (atomic accumulation target)
    const int n4 = out_size / 4;
    zero_out<<<dim3((n4 + 255) / 256), dim3(256), 0, stream>>>(out, n4);

    dim3 grid(OUT_F / NTILE, KSPLIT);   // 128 x 4 = 512 blocks
    dim3 block(256);                    // 8 wave32 waves
    density_linear_wmma<<<grid, block, 0, stream>>>(x, lw, scale, thr, step, out);
}